// GATClassifier_53755810676762
// MI455X (gfx1250) — compile-verified
//
#include <hip/hip_runtime.h>
#include <math.h>

// ---------------- problem constants (match reference) ----------------
#define IN_CH   396
#define H1      4
#define C1      32
#define HC1     128
#define H2      1
#define C2      16
#define HC2     16
#define NEG_SLOPE 0.2f

typedef float v2f __attribute__((ext_vector_type(2)));
typedef float v8f __attribute__((ext_vector_type(8)));

// ---------------------------------------------------------------------
// FP32 WMMA GEMM, register-blocked along N:
//   out[M,Nout] = A[M,K] @ W[K,Nout] + bias[Nout]
// One wave owns a 16 x (NT*16) strip: A fragment loaded once per k-step,
// reused across NT independent V_WMMA_F32_16X16X4_F32 accumulator chains.
// Requires: M%16==0, Nout==NT*16, K%4==0 (aligned float2 A loads).
// Block = 256 threads = 8 waves, each wave a different M-tile.
// ---------------------------------------------------------------------
template <int NT>
__global__ __launch_bounds__(256)
void gemm_wmma_f32(const float* __restrict__ A, const float* __restrict__ W,
                   const float* __restrict__ bias, float* __restrict__ out,
                   int Mtiles, int K) {
    const int Nout  = NT * 16;
    const int wave  = threadIdx.x >> 5;
    const int lane  = threadIdx.x & 31;
    const int mtile = blockIdx.x * 8 + wave;
    if (mtile >= Mtiles) return;              // wave-uniform: EXEC stays all-1s
    const int half = lane >> 4;               // 0: lanes 0-15, 1: lanes 16-31
    const int l16  = lane & 15;
    const int rowA = mtile * 16 + l16;

    v8f c[NT];
    #pragma unroll
    for (int t = 0; t < NT; ++t) c[t] = (v8f){};

    for (int k0 = 0; k0 < K; k0 += 4) {
        const int ka = k0 + half * 2;         // lanes 0-15: K=k0,k0+1 ; 16-31: k0+2,k0+3
        // A fragment (16x4 f32): one aligned float2 per lane, loaded ONCE per k-step
        const v2f a = *(const v2f*)(A + (size_t)rowA * K + ka);
        #pragma unroll
        for (int t = 0; t < NT; ++t) {
            const int col = t * 16 + l16;
            v2f b;                            // B fragment (4x16 f32)
            b.x = W[(size_t)(ka)     * Nout + col];
            b.y = W[(size_t)(ka + 1) * Nout + col];
            // (neg_a, A, neg_b, B, c_mod, C, reuse_a, reuse_b)
            c[t] = __builtin_amdgcn_wmma_f32_16x16x4_f32(false, a, false, b,
                                                         (short)0, c[t], false, false);
        }
    }
    #pragma unroll
    for (int t = 0; t < NT; ++t) {
        const int col  = t * 16 + l16;
        const float bc = bias[col];
        #pragma unroll
        for (int j = 0; j < 8; ++j) {
            const int r = mtile * 16 + j + half * 8;   // C/D VGPR layout
            out[(size_t)r * Nout + col] = c[t][j] + bc;
        }
    }
}

// ---------------------------------------------------------------------
// helpers
// ---------------------------------------------------------------------
__device__ __forceinline__ unsigned map_f2u(float f) {
    unsigned u = __float_as_uint(f);
    return (u & 0x80000000u) ? ~u : (u | 0x80000000u);   // order-preserving
}
__device__ __forceinline__ float unmap_u2f(unsigned u) {
    unsigned bits = (u & 0x80000000u) ? (u & 0x7FFFFFFFu) : ~u;
    return __uint_as_float(bits);
}

__global__ void zero_f(float* p, long n) {
    long i = (long)blockIdx.x * blockDim.x + threadIdx.x;
    if (i < n) p[i] = 0.0f;
}

// deg[dst] += 1 ; loop_sum[dst] += eattr  (original E edges only)
__global__ void deg_loop(const int* __restrict__ ei, const float* __restrict__ eattr,
                         float* __restrict__ deg, float* __restrict__ loop_sum, int E) {
    int e = blockIdx.x * blockDim.x + threadIdx.x;
    if (e >= E) return;
    int dst = ei[E + e];
    atomicAdd(deg + dst, 1.0f);
    atomicAdd(loop_sum + dst, eattr[e]);
}

__global__ void loop_div(float* __restrict__ loop_sum, const float* __restrict__ deg, int n) {
    int i = blockIdx.x * blockDim.x + threadIdx.x;
    if (i >= n) return;
    loop_sum[i] = loop_sum[i] / fmaxf(deg[i], 1.0f);
}

// write idx1 and idx2 output rows (both identical: [src_f ; dst_f] as floats)
__global__ void write_idx(const int* __restrict__ ei, float* __restrict__ o_idx1,
                          float* __restrict__ o_idx2, int E, int Et) {
    int e = blockIdx.x * blockDim.x + threadIdx.x;
    if (e >= Et) return;
    int src, dst;
    if (e < E) { src = ei[e]; dst = ei[E + e]; } else { src = dst = e - E; }
    float fs = (float)src, fd = (float)dst;
    o_idx1[e] = fs; o_idx1[Et + e] = fd;
    o_idx2[e] = fs; o_idx2[Et + e] = fd;
}

// score[e,h] = att[h,:] . leaky_relu(xl[src]+xr[dst]+ea*We[h,:]) ; segment max
__global__ void gat_score(const int* __restrict__ ei, const float* __restrict__ eattr,
                          const float* __restrict__ loop_attr,
                          const float* __restrict__ xl, const float* __restrict__ xr,
                          const float* __restrict__ att, const float* __restrict__ We,
                          float* __restrict__ score, unsigned* __restrict__ smax,
                          int E, int Et, int H, int C) {
    int gid = blockIdx.x * blockDim.x + threadIdx.x;
    if (gid >= Et * H) return;
    int e = gid / H, h = gid % H;
    int src, dst; float ea;
    if (e < E) { src = ei[e]; dst = ei[E + e]; ea = eattr[e]; }
    else       { src = dst = e - E;            ea = loop_attr[e - E]; }
    const int HC = H * C;
    const float* pl = xl + (size_t)src * HC + h * C;
    const float* pr = xr + (size_t)dst * HC + h * C;
    const float* pa = att + h * C;
    const float* pe = We  + h * C;
    float s = 0.0f;
    for (int c = 0; c < C; ++c) {
        float v = pl[c] + pr[c] + ea * pe[c];
        v = (v > 0.0f) ? v : NEG_SLOPE * v;
        s += pa[c] * v;
    }
    score[gid] = s;
    atomicMax(smax + (size_t)dst * H + h, map_f2u(s));
}

// ex = exp(score - smax[dst]) ; denom[dst] += ex (score overwritten with ex)
__global__ void gat_exp(const int* __restrict__ ei, float* __restrict__ score,
                        const unsigned* __restrict__ smax, float* __restrict__ denom,
                        int E, int Et, int H) {
    int gid = blockIdx.x * blockDim.x + threadIdx.x;
    if (gid >= Et * H) return;
    int e = gid / H, h = gid % H;
    int dst = (e < E) ? ei[E + e] : (e - E);
    float m  = unmap_u2f(smax[(size_t)dst * H + h]);
    float ex = expf(score[gid] - m);
    score[gid] = ex;
    atomicAdd(denom + (size_t)dst * H + h, ex);
}

// alpha = ex/denom ; acc[dst] += xl[src]*alpha ; alpha -> w output
__global__ void gat_scatter(const int* __restrict__ ei, const float* __restrict__ score,
                            const float* __restrict__ denom, const float* __restrict__ xl,
                            float* __restrict__ acc, float* __restrict__ w_out,
                            int E, int Et, int H, int C) {
    int gid = blockIdx.x * blockDim.x + threadIdx.x;
    if (gid >= Et * H) return;
    int e = gid / H, h = gid % H;
    int src, dst;
    if (e < E) { src = ei[e]; dst = ei[E + e]; } else { src = dst = e - E; }
    float alpha = score[gid] / denom[(size_t)dst * H + h];
    w_out[gid] = alpha;
    const int HC = H * C;
    const float* pl = xl  + (size_t)src * HC + h * C;
    float*       pd = acc + (size_t)dst * HC + h * C;
    for (int c = 0; c < C; ++c) atomicAdd(pd + c, pl[c] * alpha);
}

// h = elu(acc + bias)   (layer-1 output feeding layer 2)
__global__ void final_elu(const float* __restrict__ acc, const float* __restrict__ bias,
                          float* __restrict__ hbuf, long n, int HC) {
    long i = (long)blockIdx.x * blockDim.x + threadIdx.x;
    if (i >= n) return;
    float v = acc[i] + bias[i % HC];
    hbuf[i] = (v > 0.0f) ? v : expm1f(v);
}

// emb = acc + bias   (layer-2 output, H2==1 so mean over heads is identity)
__global__ void final_emb(const float* __restrict__ acc, const float* __restrict__ bias,
                          float* __restrict__ emb, long n, int HC) {
    long i = (long)blockIdx.x * blockDim.x + threadIdx.x;
    if (i >= n) return;
    emb[i] = acc[i] + bias[i % HC];
}

// ---------------------------------------------------------------------
extern "C" void kernel_launch(void* const* d_in, const int* in_sizes, int n_in,
                              void* d_out, int out_size, void* d_ws, size_t ws_size,
                              hipStream_t stream) {
    const float* x     = (const float*)d_in[0];
    const int*   ei    = (const int*)  d_in[1];
    const float* eattr = (const float*)d_in[2];
    const float* Wl1   = (const float*)d_in[3];
    const float* bl1   = (const float*)d_in[4];
    const float* Wr1   = (const float*)d_in[5];
    const float* br1   = (const float*)d_in[6];
    const float* We1   = (const float*)d_in[7];
    const float* att1  = (const float*)d_in[8];
    const float* bias1 = (const float*)d_in[9];
    const float* Wl2   = (const float*)d_in[10];
    const float* bl2   = (const float*)d_in[11];
    const float* Wr2   = (const float*)d_in[12];
    const float* br2   = (const float*)d_in[13];
    const float* We2   = (const float*)d_in[14];
    const float* att2  = (const float*)d_in[15];
    const float* bias2 = (const float*)d_in[16];

    const int N  = in_sizes[0] / IN_CH;      // 50000
    const int E  = in_sizes[2];              // 800000 (edge_attr is [E,1])
    const int Et = E + N;                    // with self-loops
    const int Mtiles = N / 16;               // 3125

    // ---- workspace layout (floats), zero-init region first ----
    float* ws = (float*)d_ws;
    size_t off = 0;
    float* deg     = ws + off; off += N;
    float* loopat  = ws + off; off += N;          // loop_sum -> loop_attr (in place)
    float* smax1f  = ws + off; off += (size_t)N * H1;
    float* den1    = ws + off; off += (size_t)N * H1;
    float* acc1    = ws + off; off += (size_t)N * HC1;
    float* smax2f  = ws + off; off += N;
    float* den2    = ws + off; off += N;
    float* acc2    = ws + off; off += (size_t)N * HC2;
    const long zero_cnt = (long)off;              // everything above re-zeroed per call
    float* xl1    = ws + off; off += (size_t)N * HC1;
    float* xr1    = ws + off; off += (size_t)N * HC1;
    float* hbuf   = ws + off; off += (size_t)N * HC1;
    float* xl2    = ws + off; off += (size_t)N * HC2;
    float* xr2    = ws + off; off += (size_t)N * HC2;
    float* score1 = ws + off; off += (size_t)Et * H1;
    float* score2 = ws + off; off += (size_t)Et * H2;
    (void)ws_size; (void)n_in; (void)out_size;

    // ---- output layout (floats) ----
    float* o_emb  = (float*)d_out;                        // [N,16]
    float* o_idx1 = o_emb  + (size_t)N * HC2;             // [2,Et]
    float* o_w1   = o_idx1 + 2 * (size_t)Et;              // [Et,4]
    float* o_idx2 = o_w1   + (size_t)Et * H1;             // [2,Et]
    float* o_w2   = o_idx2 + 2 * (size_t)Et;              // [Et,1]

    const int TB = 256;
    auto nb = [](long n, int b) { return (unsigned)((n + b - 1) / b); };

    // 0) zero accumulators (order-preserving uint map: 0u < map(any float))
    zero_f<<<nb(zero_cnt, TB), TB, 0, stream>>>(ws, zero_cnt);

    // 1) self-loop attr = mean incoming edge_attr
    deg_loop<<<nb(E, TB), TB, 0, stream>>>(ei, eattr, deg, loopat, E);
    loop_div<<<nb(N, TB), TB, 0, stream>>>(loopat, deg, N);

    // 2) layer-1 projections via fp32 WMMA (K=396, Nout=128, 8-wide N-blocking)
    {
        unsigned g = nb(Mtiles, 8);
        gemm_wmma_f32<HC1 / 16><<<g, 256, 0, stream>>>(x, Wl1, bl1, xl1, Mtiles, IN_CH);
        gemm_wmma_f32<HC1 / 16><<<g, 256, 0, stream>>>(x, Wr1, br1, xr1, Mtiles, IN_CH);
    }

    // 3) edge-index outputs (identical for both layers)
    write_idx<<<nb(Et, TB), TB, 0, stream>>>(ei, o_idx1, o_idx2, E, Et);

    // 4) layer-1 edge softmax + aggregate
    {
        long t = (long)Et * H1;
        gat_score  <<<nb(t, TB), TB, 0, stream>>>(ei, eattr, loopat, xl1, xr1, att1, We1,
                                                  score1, (unsigned*)smax1f, E, Et, H1, C1);
        gat_exp    <<<nb(t, TB), TB, 0, stream>>>(ei, score1, (const unsigned*)smax1f, den1,
                                                  E, Et, H1);
        gat_scatter<<<nb(t, TB), TB, 0, stream>>>(ei, score1, den1, xl1, acc1, o_w1,
                                                  E, Et, H1, C1);
        final_elu  <<<nb((long)N * HC1, TB), TB, 0, stream>>>(acc1, bias1, hbuf,
                                                              (long)N * HC1, HC1);
    }

    // 5) layer-2 projections via fp32 WMMA (K=128, Nout=16)
    {
        unsigned g = nb(Mtiles, 8);
        gemm_wmma_f32<HC2 / 16><<<g, 256, 0, stream>>>(hbuf, Wl2, bl2, xl2, Mtiles, HC1);
        gemm_wmma_f32<HC2 / 16><<<g, 256, 0, stream>>>(hbuf, Wr2, br2, xr2, Mtiles, HC1);
    }

    // 6) layer-2 edge softmax + aggregate, final embedding
    {
        long t = (long)Et * H2;
        gat_score  <<<nb(t, TB), TB, 0, stream>>>(ei, eattr, loopat, xl2, xr2, att2, We2,
                                                  score2, (unsigned*)smax2f, E, Et, H2, C2);
        gat_exp    <<<nb(t, TB), TB, 0, stream>>>(ei, score2, (const unsigned*)smax2f, den2,
                                                  E, Et, H2);
        gat_scatter<<<nb(t, TB), TB, 0, stream>>>(ei, score2, den2, xl2, acc2, o_w2,
                                                  E, Et, H2, C2);
        final_emb  <<<nb((long)N * HC2, TB), TB, 0, stream>>>(acc2, bias2, o_emb,
                                                              (long)N * HC2, HC2);
    }
}